// VQLoss_15539191677468
// MI455X (gfx1250) — compile-verified
//
#include <hip/hip_runtime.h>
#include <hip/hip_bf16.h>

typedef __attribute__((ext_vector_type(2))) float v2f;
typedef __attribute__((ext_vector_type(8))) float v8f;

#define B_ 4
#define Q_ 64
#define N_ 4096
#define K_ 512
#define C_ 256
#define T_ 32768

#define NA_BLOCKS 128   // VQ kernel blocks: 4 batches * 32
#define NB_BLOCKS 512   // recon kernel blocks: B*T/256
// ws float layout: [0,NA) vq partial sums | [NA,NA+NB) recon partials | [NA+NB,+4) edge minDist per batch

// ---------------------------------------------------------------------------
// Kernel A: minDist[b,n] = ||ze||^2 - 2*max_k( <ze,emb_k> - 0.5*||emb_k||^2 )
// One wave handles a 16-row (n) tile across all K=512 codes via
// V_WMMA_F32_16X16X4_F32, accumulating Q=64 in 16 steps of K=4.
// B fragments for a whole k-tile are loaded up-front (clause of 16 loads,
// single wait), then the 16-WMMA accumulation chain runs back-to-back.
// ---------------------------------------------------------------------------
__global__ __launch_bounds__(256) void vq_dist_kernel(const float* __restrict__ ze,
                                                      const float* __restrict__ emb,
                                                      float* __restrict__ ws) {
  const int lane = threadIdx.x & 31;
  const int wave = threadIdx.x >> 5;
  const int half = lane >> 4;        // 0: lanes 0-15, 1: lanes 16-31
  const int m    = lane & 15;
  const int b    = blockIdx.x >> 5;                       // 32 blocks per batch
  const int tile = ((blockIdx.x & 31) << 3) + wave;       // row tile in [0,256)
  const int n0   = tile << 4;

  const float* zeB = ze + (size_t)b * Q_ * N_;

  // Preload A fragments (ISA layout: lanes 0-15 -> K=0,1 ; lanes 16-31 -> K=2,3)
  v2f afrag[16];
  float ze2p = 0.f;
#pragma unroll
  for (int i = 0; i < 16; ++i) {
    const int qa = (i << 2) + (half << 1);
    const float x0 = zeB[(size_t)qa * N_ + n0 + m];
    const float x1 = zeB[(size_t)(qa + 1) * N_ + n0 + m];
    afrag[i].x = x0; afrag[i].y = x1;
    ze2p += x0 * x0 + x1 * x1;
  }
  // full ||ze[:,n0+m]||^2 on both lanes m and m+16 (each half saw half the q's)
  const float ze2 = ze2p + __shfl_xor(ze2p, 16, 32);

  const float2* embv = (const float2*)emb;   // emb[k*64 + q] -> 32 float2 per code row

  float smax[8];
#pragma unroll
  for (int r = 0; r < 8; ++r) smax[r] = -3.402823466e38f;

  for (int k0 = 0; k0 < K_; k0 += 16) {
    const float2* embrow = embv + (size_t)(k0 + m) * 32 + half;  // q offset 2*half

    // Stage 1: batch all 16 B-fragment loads (compiler can clause these).
    float2 bfrag[16];
#pragma unroll
    for (int i = 0; i < 16; ++i) bfrag[i] = embrow[i * 2];       // q0 = 4*i

    // Stage 2: WMMA accumulation chain + ||emb||^2 partial (co-executes).
    v8f c = {};
    float e2p = 0.f;
#pragma unroll
    for (int i = 0; i < 16; ++i) {
      v2f bf; bf.x = bfrag[i].x; bf.y = bfrag[i].y;
      e2p += bfrag[i].x * bfrag[i].x + bfrag[i].y * bfrag[i].y;
      // D = A(16x4,f32) x B(4x16,f32) + C
      c = __builtin_amdgcn_wmma_f32_16x16x4_f32(false, afrag[i], false, bf,
                                                (short)0, c, false, false);
    }
    const float e2 = e2p + __shfl_xor(e2p, 16, 32);  // full ||emb[k0+m]||^2
#pragma unroll
    for (int r = 0; r < 8; ++r)
      smax[r] = fmaxf(smax[r], c[r] - 0.5f * e2);    // column k0+m, row r+8*half
  }

  // column-max across the 16 lanes of each half (rows stay within a half)
#pragma unroll
  for (int off = 1; off < 16; off <<= 1) {
#pragma unroll
    for (int r = 0; r < 8; ++r)
      smax[r] = fmaxf(smax[r], __shfl_xor(smax[r], off, 32));
  }

  // owner lanes: half 0, m<8 owns row m (smax[m]); half 1, m>=8 owns row m (smax[m-8])
  float val = 0.f;
  const bool owner = (half == 0) ? (m < 8) : (m >= 8);
  if (owner) {
    const float sm = (half == 0) ? smax[m] : smax[m - 8];
    const float md = ze2 - 2.f * sm;     // min_k dist for n = n0+m
    val = md;
    if (n0 + m == N_ - 1) ws[NA_BLOCKS + NB_BLOCKS + b] = md;  // unique writer per b
  }

  // deterministic block reduction
#pragma unroll
  for (int off = 1; off < 32; off <<= 1) val += __shfl_xor(val, off, 32);
  __shared__ float sdata[8];
  if (lane == 0) sdata[wave] = val;
  __syncthreads();
  if (threadIdx.x == 0) {
    float s = 0.f;
#pragma unroll
    for (int w = 0; w < 8; ++w) s += sdata[w];
    ws[blockIdx.x] = s;
  }
}

// ---------------------------------------------------------------------------
// Kernel B: per (b,t): logsumexp_c(qp[b,:,t]) - qp[b,target,t], one streaming
// pass over the 134 MB quant_pred tensor (online logsumexp, coalesced in t).
// This kernel is the roofline bottleneck: 134 MB / 23.3 TB/s ~= 5.8 us.
// ---------------------------------------------------------------------------
__global__ __launch_bounds__(256) void recon_kernel(const float* __restrict__ qp,
                                                    const int* __restrict__ tw,
                                                    float* __restrict__ ws) {
  const int gid = blockIdx.x * 256 + threadIdx.x;   // [0, B*T)
  const int b = gid >> 15;                          // T = 32768
  const int t = gid & (T_ - 1);
  const float* col = qp + (size_t)b * C_ * T_ + t;
  const int target = tw[gid];

  float mval = -3.402823466e38f;
  float ssum = 0.f;
  float xt = 0.f;
#pragma unroll 4
  for (int c = 0; c < C_; ++c) {
    const float x = col[(size_t)c * T_];
    xt = (c == target) ? x : xt;
    const float mnew = fmaxf(mval, x);
    ssum = ssum * __expf(mval - mnew) + __expf(x - mnew);
    mval = mnew;
  }
  float val = mval + __logf(ssum) - xt;

#pragma unroll
  for (int off = 1; off < 32; off <<= 1) val += __shfl_xor(val, off, 32);
  __shared__ float sdata[8];
  const int lane = threadIdx.x & 31;
  const int wave = threadIdx.x >> 5;
  if (lane == 0) sdata[wave] = val;
  __syncthreads();
  if (threadIdx.x == 0) {
    float s = 0.f;
#pragma unroll
    for (int w = 0; w < 8; ++w) s += sdata[w];
    ws[NA_BLOCKS + blockIdx.x] = s;
  }
}

// ---------------------------------------------------------------------------
// Final: sum partials in fixed order; conv-transpose collapsed analytically:
// sum_t conv(x)[0..T) = (sum_j w[j]) * sum_n x[n]  -  w[0] * x[N-1]
// total = ( rec + (1+gamma) * (W*S_dist - w0*S_edge) ) / (B*T)
// ---------------------------------------------------------------------------
__global__ void finalize_kernel(const float* __restrict__ ws,
                                const float* __restrict__ cw,
                                float* __restrict__ out) {
  float sd = 0.f;
  for (int i = 0; i < NA_BLOCKS; ++i) sd += ws[i];
  float sr = 0.f;
  for (int i = 0; i < NB_BLOCKS; ++i) sr += ws[NA_BLOCKS + i];
  float se = 0.f;
  for (int b = 0; b < B_; ++b) se += ws[NA_BLOCKS + NB_BLOCKS + b];
  float W = 0.f;
  for (int j = 0; j < 9; ++j) W += cw[j];
  const float total = sr + 1.25f * (W * sd - cw[0] * se);
  out[0] = total / (float)(B_ * T_);
}

extern "C" void kernel_launch(void* const* d_in, const int* in_sizes, int n_in,
                              void* d_out, int out_size, void* d_ws, size_t ws_size,
                              hipStream_t stream) {
  const float* qp  = (const float*)d_in[0];  // (B,C,T)
  const int*   tw  = (const int*)  d_in[1];  // (B,T)
  const float* ze  = (const float*)d_in[2];  // (B,Q,N)
  const float* emb = (const float*)d_in[3];  // (K,Q)
  const float* cw  = (const float*)d_in[4];  // (9,)
  float* out = (float*)d_out;
  float* ws  = (float*)d_ws;

  vq_dist_kernel<<<NA_BLOCKS, 256, 0, stream>>>(ze, emb, ws);
  recon_kernel<<<NB_BLOCKS, 256, 0, stream>>>(qp, tw, ws);
  finalize_kernel<<<1, 1, 0, stream>>>(ws, cw, out);
}